// DenseFilterExpansion_22127671509395
// MI455X (gfx1250) — compile-verified
//
#include <hip/hip_runtime.h>
#include <hip/hip_bf16.h>

// out[b, f, s] = x[b, s] * w[f, s] + bias[f, s]
// BATCH=128, NUM_FILTERS=256, SEQ_LEN=4096, all f32.
// Pure store-bandwidth kernel: 512MB written once -> ~22us floor at 23.3 TB/s.

typedef float        v4f __attribute__((ext_vector_type(4)));
typedef unsigned int u32;
typedef u32          v4u __attribute__((ext_vector_type(4)));
typedef int          v4i __attribute__((ext_vector_type(4)));
typedef int          v8i __attribute__((ext_vector_type(8)));

#define SEQ   4096
#define SEQ4  (SEQ / 4)       // 1024 float4 per row
#define NF    256
#define NB    128
#define BPB   8               // batches per block (register-blocked)
#define TPB   256             // threads per block (8 waves, wave32)

// ---------------------------------------------------------------------------
// TDM: DMA one contiguous f32 row (SEQ elements = 16KB) from global to LDS.
// Descriptor per CDNA5 ISA ch.8:
//   group0: [count=1 | lds_addr | global_addr(57b) | type=2]
//   group1: data_size=2 (4B), tensor_dim0=SEQ, tensor_dim1=1,
//           tile_dim0=SEQ, tile_dim1=1, tensor_dim0_stride=SEQ
// ---------------------------------------------------------------------------
__device__ __forceinline__ void tdm_load_row_f32(const float* gsrc, u32 lds_byte_off) {
  unsigned long long ga = (unsigned long long)(size_t)gsrc;

  v4u g0;
  g0.x = 1u;                                            // count=1, user mode, no gather
  g0.y = lds_byte_off;                                  // LDS destination (bytes)
  g0.z = (u32)(ga & 0xFFFFFFFFull);                     // global_addr[31:0]
  g0.w = (u32)((ga >> 32) & 0x01FFFFFFull)              // global_addr[56:32]
       | (2u << 30);                                    // type = 2 ("image")

  v8i g1;
  g1[0] = (int)(2u << 16);                              // workgroup_mask=0, data_size=2 (4B)
  g1[1] = (int)((u32)SEQ << 16);                        // tensor_dim0[15:0] in [31:16]
  g1[2] = (int)(1u << 16);                              // tensor_dim0[31:16]=0, tensor_dim1=1
  g1[3] = (int)((u32)SEQ << 16);                        // tensor_dim1 hi=0, tile_dim0=SEQ
  g1[4] = 1;                                            // tile_dim1=1, tile_dim2=0
  g1[5] = SEQ;                                          // tensor_dim0_stride[31:0]
  g1[6] = 0;                                            // stride hi, dim1_stride lo
  g1[7] = 0;

  v4i z4 = 0;
#if defined(__clang_major__) && (__clang_major__ >= 23)
  v8i z8 = 0;
  __builtin_amdgcn_tensor_load_to_lds(g0, g1, z4, z4, z8, 0);   // 6-arg toolchain
#else
  __builtin_amdgcn_tensor_load_to_lds(g0, g1, z4, z4, 0);       // 5-arg ROCm 7.2
#endif
}

__global__ __launch_bounds__(TPB) void DenseFilterExpansion_kernel(
    const float* __restrict__ x,     // [NB, 1, SEQ]
    const float* __restrict__ w,     // [NF, SEQ]
    const float* __restrict__ bias,  // [NF, SEQ]
    float*       __restrict__ out)   // [NB, NF, SEQ]
{
  __shared__ v4f s_w[SEQ4];   // 16KB: w[f, :]
  __shared__ v4f s_b[SEQ4];   // 16KB: bias[f, :]

  const int f  = (int)blockIdx.x;         // filter
  const int bg = (int)blockIdx.y * BPB;   // first batch of this block's group

  // Wave 0 issues two TDM row loads into LDS, waits for TENSORcnt==0,
  // then everyone syncs at the workgroup barrier.
  if (threadIdx.x == 0) {
    tdm_load_row_f32(w    + (size_t)f * SEQ, (u32)(size_t)(void*)s_w);
    tdm_load_row_f32(bias + (size_t)f * SEQ, (u32)(size_t)(void*)s_b);
    __builtin_amdgcn_s_wait_tensorcnt(0);
  }
  __syncthreads();

  const v4f* x4 = (const v4f*)x + (size_t)bg * SEQ4;  // batch group's x rows
  v4f*       o4 = (v4f*)out;
  const size_t obase = ((size_t)bg * NF + (size_t)f) * SEQ4;

#pragma unroll
  for (int c = 0; c < SEQ4 / TPB; ++c) {              // 4 seq chunks
    const int s4 = c * TPB + (int)threadIdx.x;
    const v4f wv = s_w[s4];                           // reused across BPB batches
    const v4f bv = s_b[s4];
#pragma unroll
    for (int i = 0; i < BPB; ++i) {
      const v4f xv = x4[(size_t)i * SEQ4 + s4];       // L2-resident (x is 2MB)
      const v4f ov = xv * wv + bv;                    // v_fma / v_pk_fma
      // Streaming output never re-read: nontemporal b128 store.
      __builtin_nontemporal_store(ov, o4 + obase + (size_t)i * ((size_t)NF * SEQ4) + s4);
    }
  }
}

extern "C" void kernel_launch(void* const* d_in, const int* in_sizes, int n_in,
                              void* d_out, int out_size, void* d_ws, size_t ws_size,
                              hipStream_t stream) {
  const float* x    = (const float*)d_in[0];  // (128, 1, 4096) f32
  const float* w    = (const float*)d_in[1];  // (256, 4096)    f32
  const float* bias = (const float*)d_in[2];  // (256, 4096)    f32
  float*       out  = (float*)d_out;          // (128, 256, 4096) f32

  dim3 grid(NF, NB / BPB);  // 256 x 16 = 4096 blocks
  DenseFilterExpansion_kernel<<<grid, TPB, 0, stream>>>(x, w, bias, out);
}